// SupConLoss_5866925326682
// MI455X (gfx1250) — compile-verified
//
#include <hip/hip_runtime.h>
#include <math.h>

// SupCon loss, B=8192, D=128, fp32.
// GEMM = F*F^T via V_WMMA_F32_16X16X4_F32 (fp32 WMMA), flash-style streaming
// softmax stats, double-buffered LDS staging via gfx1250 async global->LDS
// copies (ASYNCcnt), dual WMMA accumulator chains, deterministic reduction.

#define BSZ 8192
#define DIM 128
#define COLS_PER_STAGE 32
#define NSTAGE (BSZ / COLS_PER_STAGE)   // 256 staging iterations
#define ROWS_PER_BLOCK 128              // 8 waves * 16 rows
#define NBLOCKS (BSZ / ROWS_PER_BLOCK)  // 64
#define LDS_STRIDE 132                  // 128 + 4 pad floats -> conflict-free b64 reads

#if __has_builtin(__builtin_amdgcn_global_load_async_to_lds_b128)
#define USE_ASYNC 1
#else
#define USE_ASYNC 0
#endif

typedef float v2f __attribute__((ext_vector_type(2)));
typedef float v8f __attribute__((ext_vector_type(8)));
typedef int   v4i __attribute__((ext_vector_type(4)));
typedef __attribute__((address_space(1))) v4i GV4;   // global int4
typedef __attribute__((address_space(3))) v4i LV4;   // LDS int4

__global__ __launch_bounds__(256)
void supcon_main(const float* __restrict__ F,
                 const int* __restrict__ labels,
                 const int* __restrict__ scale_w,
                 const int* __restrict__ fac_lab,
                 float* __restrict__ partials)
{
    __shared__ float ldsB[2][COLS_PER_STAGE * LDS_STRIDE];
    __shared__ float wavePartial[16];

    const int t    = threadIdx.x;
    const int wave = t >> 5;        // 8 waves (wave32)
    const int lane = t & 31;
    const int half = lane >> 4;     // 0: rows M=0..7 / K lo ; 1: rows M=8..15 / K hi
    const int l16  = lane & 15;

    const float invT = 1.0f / 0.07f;

    const int rowTileBase = blockIdx.x * ROWS_PER_BLOCK + wave * 16;
    const int myRow = rowTileBase + l16;   // A-matrix M index for this lane

    // Preload A fragments for all 32 K-steps: lane needs F[myRow][4*kk + 2*half + {0,1}]
    v2f afrag[32];
    #pragma unroll
    for (int kk = 0; kk < 32; ++kk)
        afrag[kk] = *(const v2f*)(F + myRow * DIM + 4 * kk + 2 * half);

    // Labels of the 8 rows this lane accumulates stats for (C layout: M = 8*half + r)
    int rowlab[8];
    #pragma unroll
    for (int r = 0; r < 8; ++r)
        rowlab[r] = labels[rowTileBase + 8 * half + r];

    // Per-lane streaming softmax stats over this lane's column subset
    float m[8], S[8], sumL[8], cnt[8];
    #pragma unroll
    for (int r = 0; r < 8; ++r) { m[r] = -INFINITY; S[r] = 0.f; sumL[r] = 0.f; cnt[r] = 0.f; }

    // Staging role: thread t handles column (stage*32 + sn), dims [sk, sk+16)
    const int sn = t >> 3;          // 0..31
    const int sk = (t & 7) * 16;    // 0,16,...,112

#if USE_ASYNC
    // Async stage: 4x b128 per lane, memory -> LDS, no VGPR data path.
    auto stage_async = [&](int stage, int buf) {
        const float* g = F + (stage * COLS_PER_STAGE + sn) * DIM + sk;
        float* l = &ldsB[buf][sn * LDS_STRIDE + sk];
        #pragma unroll
        for (int q = 0; q < 4; ++q)
            __builtin_amdgcn_global_load_async_to_lds_b128(
                (GV4*)(g + 4 * q), (LV4*)(l + 4 * q), 0, 0);
    };

    stage_async(0, 0);
    __builtin_amdgcn_s_wait_asynccnt(0);
    __syncthreads();               // buffer 0 visible to all waves
#endif

    for (int jj = 0; jj < NSTAGE; ++jj) {
#if USE_ASYNC
        if (jj + 1 < NSTAGE) stage_async(jj + 1, (jj + 1) & 1);  // overlaps compute
        const float* curbuf = ldsB[jj & 1];
#else
        // Fallback: synchronous stage, no long-lived prefetch registers (no spill)
        __syncthreads();
        {
            const float4* src = (const float4*)(F + (jj * COLS_PER_STAGE + sn) * DIM + sk);
            float* dst = ldsB[0] + sn * LDS_STRIDE + sk;
            #pragma unroll
            for (int q = 0; q < 4; ++q) ((float4*)dst)[q] = src[q];
        }
        __syncthreads();
        const float* curbuf = ldsB[0];
#endif

        #pragma unroll
        for (int sub = 0; sub < 2; ++sub) {
            const int  colbase = jj * COLS_PER_STAGE + sub * 16;
            const int  coln = colbase + l16;      // C layout: N = lane%16
            const int  lcol = labels[coln];

            v8f c0 = {}, c1 = {};
            const float* bbase = curbuf + (sub * 16 + l16) * LDS_STRIDE + 2 * half;
            #pragma unroll
            for (int kk = 0; kk < 32; kk += 2) {   // two interleaved accumulator chains
                v2f b0 = *(const v2f*)(bbase + 4 * kk);
                v2f b1 = *(const v2f*)(bbase + 4 * (kk + 1));
                c0 = __builtin_amdgcn_wmma_f32_16x16x4_f32(
                        false, afrag[kk],     false, b0, (short)0, c0, false, false);
                c1 = __builtin_amdgcn_wmma_f32_16x16x4_f32(
                        false, afrag[kk + 1], false, b1, (short)0, c1, false, false);
            }
            v8f c = c0 + c1;

            #pragma unroll
            for (int r = 0; r < 8; ++r) {
                const int row = rowTileBase + 8 * half + r;
                const float l = c[r] * invT;
                const float offdiag = (row != coln) ? 1.0f : 0.0f;
                if (l > m[r]) {                   // new running max; rescale S
                    S[r] = S[r] * __expf(m[r] - l) + offdiag;
                    m[r] = l;
                } else {
                    S[r] += offdiag * __expf(l - m[r]);
                }
                if ((rowlab[r] == lcol) && (row != coln)) {
                    sumL[r] += l;
                    cnt[r]  += 1.0f;
                }
            }
        }

#if USE_ASYNC
        if (jj + 1 < NSTAGE) __builtin_amdgcn_s_wait_asynccnt(0);
        __syncthreads();           // next buffer visible to all waves
#endif
    }

    // Merge the 16 lanes that share each row (xor shuffle within 16-lane half)
    #pragma unroll
    for (int r = 0; r < 8; ++r) {
        #pragma unroll
        for (int off = 1; off < 16; off <<= 1) {
            float om  = __shfl_xor(m[r],    off, 32);
            float oS  = __shfl_xor(S[r],    off, 32);
            float osl = __shfl_xor(sumL[r], off, 32);
            float oc  = __shfl_xor(cnt[r],  off, 32);
            float nm  = fmaxf(m[r], om);
            S[r]   = S[r] * __expf(m[r] - nm) + oS * __expf(om - nm);
            m[r]   = nm;
            sumL[r] += osl;
            cnt[r]  += oc;
        }
    }

    // Per-row loss: -(T/baseT=1) * w * (sumL - cnt*logsumexp) / cnt
    const int sw = *scale_w;
    const int fl = *fac_lab;
    float acc = 0.f;
    #pragma unroll
    for (int r = 0; r < 8; ++r) {
        const float lse = m[r] + __logf(S[r]);
        const float w   = (rowlab[r] == fl) ? (float)sw : 1.0f;
        acc += (cnt[r] > 0.f) ? (-w * (sumL[r] - cnt[r] * lse) / cnt[r]) : 0.f;
    }
    if (l16 == 0) wavePartial[wave * 2 + half] = acc;  // 16 half-wave partials
    __syncthreads();
    if (t == 0) {
        float s = 0.f;
        #pragma unroll
        for (int i = 0; i < 16; ++i) s += wavePartial[i];
        partials[blockIdx.x] = s;
    }
}

__global__ void supcon_reduce(const float* __restrict__ partials,
                              float* __restrict__ out)
{
    if (threadIdx.x == 0) {
        float s = 0.f;
        for (int i = 0; i < NBLOCKS; ++i) s += partials[i];  // fixed order: deterministic
        out[0] = s / (float)BSZ;
    }
}

extern "C" void kernel_launch(void* const* d_in, const int* in_sizes, int n_in,
                              void* d_out, int out_size, void* d_ws, size_t ws_size,
                              hipStream_t stream)
{
    const float* F      = (const float*)d_in[0];
    const int*   labels = (const int*)d_in[1];
    const int*   sw     = (const int*)d_in[2];
    const int*   fl     = (const int*)d_in[3];
    float* partials = (float*)d_ws;

    supcon_main<<<NBLOCKS, 256, 0, stream>>>(F, labels, sw, fl, partials);
    supcon_reduce<<<1, 32, 0, stream>>>(partials, (float*)d_out);
}